// GATFraudDetector_18863496364415
// MI455X (gfx1250) — compile-verified
//
#include <hip/hip_runtime.h>
#include <hip/hip_bf16.h>

// ---------------------------------------------------------------------------
// GAT fraud detector, 2 GATConv layers + linear head, fp32 throughout.
// gfx1250 (CDNA5): wave32, WMMA f32 16x16x4 tiles, native f32 global atomics.
// GEMM: one wave -> 16x64 output tile (4 WMMA accumulators share one A frag).
// Edge softmax/aggregation: 3 atomic passes; layer-1 working set (~106 MB)
// fits the 192 MB global L2, so scatter/gather runs at L2 bandwidth.
// ---------------------------------------------------------------------------

typedef float v2f __attribute__((ext_vector_type(2)));
typedef float v8f __attribute__((ext_vector_type(8)));

#define NEG_SLOPE 0.2f

// ---------------------- utility kernels ------------------------------------

__global__ void k_fill(float* __restrict__ p, float v, int n) {
  int i = blockIdx.x * blockDim.x + threadIdx.x;
  if (i < n) p[i] = v;
}

__device__ inline void atomicMaxF(float* addr, float val) {
  unsigned int old = __float_as_uint(*addr);
  while (__uint_as_float(old) < val) {
    unsigned int assumed = old;
    old = atomicCAS((unsigned int*)addr, assumed, __float_as_uint(val));
    if (old == assumed) break;
  }
}

// edge i in [0, E+N): first E from edge_index, last N are self loops
__device__ inline void edge_sd(int i, const int* __restrict__ ei, int E,
                               int* s, int* d) {
  if (i < E) { *s = ei[i]; *d = ei[E + i]; }
  else       { *s = i - E; *d = i - E; }
}

// ---------------------- WMMA GEMM (fp32) -----------------------------------
// C[M,N] = A[M,K] @ B[K,N], row-major, fp32. N must be a multiple of 64.
// grid = (ceil(M/16), N/64), block = 32 (one wave -> one 16x64 tile strip).
// Fragment layouts per CDNA5 ISA 7.12.2:
//   A 16x4 : lanes 0-15 M=0..15 ; vgpr0 = K {0 | 2}, vgpr1 = K {1 | 3}
//   B 4x16 : lanes -> N ; vgpr0 = K {0 | 2}, vgpr1 = K {1 | 3}
//   C/D    : vgpr r : lanes 0-15 -> M=r, lanes 16-31 -> M=r+8
__global__ void k_gemm_wmma_f32(const float* __restrict__ A,
                                const float* __restrict__ B,
                                float* __restrict__ C,
                                int M, int N, int K) {
  const int lane = threadIdx.x;
  const int lo = lane & 15;
  const int hi = lane >> 4;
  const int m0 = blockIdx.x * 16;
  const int n0 = blockIdx.y * 64;

  int mrow = m0 + lo;
  if (mrow >= M) mrow = M - 1;      // clamp: keeps EXEC all-ones for WMMA
  const float* __restrict__ Arow = A + (size_t)mrow * K;

  v8f acc0 = {}, acc1 = {}, acc2 = {}, acc3 = {};

  for (int kb = 0; kb < K; kb += 4) {
    v2f a = *(const v2f*)(Arow + kb + 2 * hi);
    const float* __restrict__ B0 = B + (size_t)(kb + 2 * hi    ) * N + n0 + lo;
    const float* __restrict__ B1 = B + (size_t)(kb + 2 * hi + 1) * N + n0 + lo;
    v2f b0 = {B0[ 0], B1[ 0]};
    v2f b1 = {B0[16], B1[16]};
    v2f b2 = {B0[32], B1[32]};
    v2f b3 = {B0[48], B1[48]};
    acc0 = __builtin_amdgcn_wmma_f32_16x16x4_f32(false, a, false, b0, (short)0, acc0, false, false);
    acc1 = __builtin_amdgcn_wmma_f32_16x16x4_f32(false, a, false, b1, (short)0, acc1, false, false);
    acc2 = __builtin_amdgcn_wmma_f32_16x16x4_f32(false, a, false, b2, (short)0, acc2, false, false);
    acc3 = __builtin_amdgcn_wmma_f32_16x16x4_f32(false, a, false, b3, (short)0, acc3, false, false);
  }

  for (int r = 0; r < 8; ++r) {
    int row = m0 + r + 8 * hi;
    if (row < M) {
      float* __restrict__ Crow = C + (size_t)row * N + n0 + lo;
      Crow[ 0] = acc0[r];
      Crow[16] = acc1[r];
      Crow[32] = acc2[r];
      Crow[48] = acc3[r];
    }
  }
}

// ---------------------- attention scores -----------------------------------
// e_src[n,h] = sum_c h[n,h,c]*a_src[h,c] ; e_dst likewise. Thread per (n,h).
__global__ void k_scores(const float* __restrict__ h,
                         const float* __restrict__ a_src,
                         const float* __restrict__ a_dst,
                         float* __restrict__ es, float* __restrict__ ed,
                         int N, int H, int C) {
  int i = blockIdx.x * blockDim.x + threadIdx.x;
  if (i >= N * H) return;
  int n = i / H, hh = i % H;
  const float* hv = h + (size_t)n * H * C + (size_t)hh * C;
  const float* as = a_src + hh * C;
  const float* ad = a_dst + hh * C;
  float s = 0.f, d = 0.f;
  for (int c = 0; c < C; ++c) { s += hv[c] * as[c]; d += hv[c] * ad[c]; }
  es[i] = s;
  ed[i] = d;
}

// ---------------------- segment softmax: pass 1 (max) ----------------------
__global__ void k_edge_max(const int* __restrict__ ei, int E, int Etot, int H,
                           const float* __restrict__ es,
                           const float* __restrict__ ed,
                           float* __restrict__ m) {
  int i = blockIdx.x * blockDim.x + threadIdx.x;
  if (i >= Etot * H) return;
  int e = i / H, hh = i - e * H;
  int s, d; edge_sd(e, ei, E, &s, &d);
  float v = es[s * H + hh] + ed[d * H + hh];
  v = v > 0.f ? v : NEG_SLOPE * v;
  atomicMaxF(&m[d * H + hh], v);
}

// ---------------------- segment softmax: pass 2 (denominator) --------------
__global__ void k_edge_denom(const int* __restrict__ ei, int E, int Etot, int H,
                             const float* __restrict__ es,
                             const float* __restrict__ ed,
                             const float* __restrict__ m,
                             float* __restrict__ dn) {
  int i = blockIdx.x * blockDim.x + threadIdx.x;
  if (i >= Etot * H) return;
  int e = i / H, hh = i - e * H;
  int s, d; edge_sd(e, ei, E, &s, &d);
  float v = es[s * H + hh] + ed[d * H + hh];
  v = v > 0.f ? v : NEG_SLOPE * v;
  atomicAdd(&dn[d * H + hh], expf(v - m[d * H + hh]));
}

// ---------------------- softmax pass 3 + message aggregation ---------------
// One block per edge; blockDim = H*C; thread t covers channel t of h[src].
// Note: with C=64, all 32 lanes of a wave share the same head -> the
// es/ed/m/dn loads are wave-uniform addresses (single cacheline hits).
__global__ void k_edge_agg(const int* __restrict__ ei, int E, int H, int C,
                           const float* __restrict__ hfeat,
                           const float* __restrict__ es,
                           const float* __restrict__ ed,
                           const float* __restrict__ m,
                           const float* __restrict__ dn,
                           float* __restrict__ out) {
  int e = blockIdx.x;
  int t = threadIdx.x;
  int hh = t / C;
  int s, d; edge_sd(e, ei, E, &s, &d);
  float msg = hfeat[(size_t)s * H * C + t];        // issue gather early
  float v = es[s * H + hh] + ed[d * H + hh];
  v = v > 0.f ? v : NEG_SLOPE * v;
  float alpha = expf(v - m[d * H + hh]) / (dn[d * H + hh] + 1e-16f);
  atomicAdd(&out[(size_t)d * H * C + t], msg * alpha);
}

// ---------------------- bias + relu ----------------------------------------
__global__ void k_bias_relu(float* __restrict__ x, const float* __restrict__ b,
                            int total, int F) {
  int i = blockIdx.x * blockDim.x + threadIdx.x;
  if (i >= total) return;
  float v = x[i] + b[i % F];
  x[i] = v > 0.f ? v : 0.f;
}

// ---------------------- output head ----------------------------------------
__global__ void k_head(const float* __restrict__ h, const float* __restrict__ w,
                       const float* __restrict__ b, float* __restrict__ o,
                       int N, int C) {
  int n = blockIdx.x * blockDim.x + threadIdx.x;
  if (n >= N) return;
  float s = b[0];
  const float* hv = h + (size_t)n * C;
  for (int c = 0; c < C; ++c) s += hv[c] * w[c];
  o[n] = s;
}

// ---------------------------------------------------------------------------

extern "C" void kernel_launch(void* const* d_in, const int* in_sizes, int n_in,
                              void* d_out, int out_size, void* d_ws, size_t ws_size,
                              hipStream_t stream) {
  const float* x    = (const float*)d_in[0];
  const int*   ei   = (const int*)  d_in[1];
  const float* W1   = (const float*)d_in[2];
  const float* as1  = (const float*)d_in[3];
  const float* ad1  = (const float*)d_in[4];
  const float* b1   = (const float*)d_in[5];
  const float* W2   = (const float*)d_in[6];
  const float* as2  = (const float*)d_in[7];
  const float* ad2  = (const float*)d_in[8];
  const float* b2   = (const float*)d_in[9];
  const float* Wout = (const float*)d_in[10];
  const float* bout = (const float*)d_in[11];

  const int FIN = 128, H1 = 4, C = 64;
  const int F1 = H1 * C;                  // 256
  const int N    = in_sizes[0] / FIN;     // 50000
  const int E    = in_sizes[1] / 2;       // 1600000
  const int Etot = E + N;                 // + self loops

  // ---- workspace layout (floats) ----
  // persistent: h1 [N*F1], out1 [N*F1], layer-1 scalars [4*N*H1]
  // layer-2 temps reuse the h1 region (N*(64+64+4) < N*256).
  float* h1   = (float*)d_ws;
  float* out1 = h1   + (size_t)N * F1;
  float* es1  = out1 + (size_t)N * F1;
  float* ed1  = es1  + (size_t)N * H1;
  float* m1   = ed1  + (size_t)N * H1;
  float* dn1  = m1   + (size_t)N * H1;

  float* h2   = h1;                       // N*C
  float* out2 = h2   + (size_t)N * C;     // N*C
  float* es2  = out2 + (size_t)N * C;     // N
  float* ed2  = es2  + N;
  float* m2   = ed2  + N;
  float* dn2  = m2   + N;

  float* logits = (float*)d_out;

  const int T = 256;
  auto blocks = [](long long n, int t) { return (unsigned)((n + t - 1) / t); };

  // =================== layer 1 (H=4, C=64) ===================
  // h1 = x @ W1
  {
    dim3 g(blocks(N, 16), F1 / 64);
    k_gemm_wmma_f32<<<g, 32, 0, stream>>>(x, W1, h1, N, F1, FIN);
  }
  k_scores<<<blocks((long long)N * H1, T), T, 0, stream>>>(h1, as1, ad1, es1, ed1, N, H1, C);

  k_fill<<<blocks((long long)N * H1, T), T, 0, stream>>>(m1, -1e30f, N * H1);
  k_fill<<<blocks((long long)N * H1, T), T, 0, stream>>>(dn1, 0.f, N * H1);
  k_fill<<<blocks((long long)N * F1, T), T, 0, stream>>>(out1, 0.f, N * F1);

  k_edge_max  <<<blocks((long long)Etot * H1, T), T, 0, stream>>>(ei, E, Etot, H1, es1, ed1, m1);
  k_edge_denom<<<blocks((long long)Etot * H1, T), T, 0, stream>>>(ei, E, Etot, H1, es1, ed1, m1, dn1);
  k_edge_agg  <<<Etot, H1 * C, 0, stream>>>(ei, E, H1, C, h1, es1, ed1, m1, dn1, out1);

  k_bias_relu<<<blocks((long long)N * F1, T), T, 0, stream>>>(out1, b1, N * F1, F1);

  // =================== layer 2 (H=1, C=64) ===================
  // h2 = out1 @ W2
  {
    dim3 g(blocks(N, 16), C / 64);
    k_gemm_wmma_f32<<<g, 32, 0, stream>>>(out1, W2, h2, N, C, F1);
  }
  k_scores<<<blocks(N, T), T, 0, stream>>>(h2, as2, ad2, es2, ed2, N, 1, C);

  k_fill<<<blocks(N, T), T, 0, stream>>>(m2, -1e30f, N);
  k_fill<<<blocks(N, T), T, 0, stream>>>(dn2, 0.f, N);
  k_fill<<<blocks((long long)N * C, T), T, 0, stream>>>(out2, 0.f, N * C);

  k_edge_max  <<<blocks(Etot, T), T, 0, stream>>>(ei, E, Etot, 1, es2, ed2, m2);
  k_edge_denom<<<blocks(Etot, T), T, 0, stream>>>(ei, E, Etot, 1, es2, ed2, m2, dn2);
  k_edge_agg  <<<Etot, C, 0, stream>>>(ei, E, 1, C, h2, es2, ed2, m2, dn2, out2);

  k_bias_relu<<<blocks((long long)N * C, T), T, 0, stream>>>(out2, b2, N * C, C);

  // =================== output head ===================
  k_head<<<blocks(N, T), T, 0, stream>>>(out2, Wout, bout, logits, N, C);
}